// MVG_19851338842311
// MI455X (gfx1250) — compile-verified
//
#include <hip/hip_runtime.h>

typedef __attribute__((ext_vector_type(2))) float v2f;
typedef __attribute__((ext_vector_type(8))) float v8f;

#define TM 128          // block tile rows (M)
#define TN 128          // block tile cols (N)
#define KSTEP 16        // K per staging step (4 wmma k-substeps of 4)
#define ASTRIDE 18      // padded LDS row stride for A tile (floats)

// out[MxN] = A[MxK] @ W[KxN] + bias[N], W = M + e^{u/2} eps e^{v/2}, bias = row K
__global__ __launch_bounds__(256) void mvg_layer_kernel(
    const float* __restrict__ A,    // [8192, K]
    const float* __restrict__ Mw,   // [K+1, N]
    const float* __restrict__ u,    // [K+1]
    const float* __restrict__ v,    // [N]
    const float* __restrict__ eps,  // [K+1, N]
    float* __restrict__ out,        // [8192, N]
    int K, int N, int relu)
{
    __shared__ float As[TM * ASTRIDE];   // 128 x 16, padded rows
    __shared__ float Ws[KSTEP * TN];     // K-pair interleaved: [(k>>1)*256 + n*2 + (k&1)]
    __shared__ float svs[TN];            // e^{v/2} for this column tile
    __shared__ float bias[TN];           // fused ones-column bias row

    const int tid  = threadIdx.x;
    const int lane = tid & 31;
    const int wave = tid >> 5;           // 8 waves
    const int wm   = wave & 1;           // 2 wave-rows  x 64
    const int wn   = wave >> 1;          // 4 wave-cols  x 32
    const int la   = lane & 15;
    const int lh   = lane >> 4;          // 0: K pair {0,1}, 1: K pair {2,3}

    const int n0 = blockIdx.x * TN;
    const int m0 = blockIdx.y * TM;

    // staging assignments
    const int ar = tid >> 1;             // A tile row 0..127
    const int ac = (tid & 1) * 8;        // A tile col 0 or 8
    const int wp = tid >> 5;             // W K-pair 0..7
    const int wc = (tid & 31) * 4;       // W col 0..124

    // column scales + bias row (weight row K == the appended-ones row)
    if (tid < TN) {
        float sv = __expf(0.5f * v[n0 + tid]);
        svs[tid] = sv;
        size_t off = (size_t)K * N + (size_t)(n0 + tid);
        bias[tid] = Mw[off] + __expf(0.5f * u[K]) * eps[off] * sv;
    }
    __syncthreads();

    // loop-invariant column scales for this thread's W slice
    const float4 svr = *(const float4*)&svs[wc];

    v8f acc[4][2];
    #pragma unroll
    for (int mi = 0; mi < 4; ++mi)
        #pragma unroll
        for (int ni = 0; ni < 2; ++ni)
            #pragma unroll
            for (int e = 0; e < 8; ++e)
                acc[mi][ni][e] = 0.0f;

    // -------- software pipeline: register-staged global prefetch --------
    float4 pa0, pa1;                 // A tile piece (row ar, cols ac..ac+7)
    float4 pm0, pm1, pe0, pe1;       // W K-pair rows (2wp, 2wp+1), cols wc..wc+3
    float  su0, su1;

    {
        const float4* gp = (const float4*)(A + (size_t)(m0 + ar) * K + (size_t)ac);
        pa0 = gp[0]; pa1 = gp[1];
        size_t off = (size_t)(2 * wp) * N + (size_t)(n0 + wc);
        pm0 = *(const float4*)(Mw + off);
        pm1 = *(const float4*)(Mw + off + N);
        pe0 = *(const float4*)(eps + off);
        pe1 = *(const float4*)(eps + off + N);
        su0 = __expf(0.5f * u[2 * wp]);
        su1 = __expf(0.5f * u[2 * wp + 1]);
    }

    for (int k0 = 0; k0 < K; k0 += KSTEP) {
        // ---- commit staged registers to LDS ----
        {
            float* dst = &As[ar * ASTRIDE + ac];
            ((float2*)dst)[0] = make_float2(pa0.x, pa0.y);
            ((float2*)dst)[1] = make_float2(pa0.z, pa0.w);
            ((float2*)dst)[2] = make_float2(pa1.x, pa1.y);
            ((float2*)dst)[3] = make_float2(pa1.z, pa1.w);

            const float sv[4] = { svr.x, svr.y, svr.z, svr.w };
            const float m0v[4] = { pm0.x, pm0.y, pm0.z, pm0.w };
            const float m1v[4] = { pm1.x, pm1.y, pm1.z, pm1.w };
            const float e0v[4] = { pe0.x, pe0.y, pe0.z, pe0.w };
            const float e1v[4] = { pe1.x, pe1.y, pe1.z, pe1.w };
            #pragma unroll
            for (int i = 0; i < 4; ++i) {
                float w0 = fmaf(su0 * e0v[i], sv[i], m0v[i]);
                float w1 = fmaf(su1 * e1v[i], sv[i], m1v[i]);
                *(float2*)&Ws[wp * 256 + (wc + i) * 2] = make_float2(w0, w1);
            }
        }
        __syncthreads();

        // ---- prefetch next K-step while wmma burst runs ----
        if (k0 + KSTEP < K) {
            const float4* gp = (const float4*)(A + (size_t)(m0 + ar) * K + (size_t)(k0 + KSTEP + ac));
            pa0 = gp[0]; pa1 = gp[1];
            size_t off = (size_t)(k0 + KSTEP + 2 * wp) * N + (size_t)(n0 + wc);
            pm0 = *(const float4*)(Mw + off);
            pm1 = *(const float4*)(Mw + off + N);
            pe0 = *(const float4*)(eps + off);
            pe1 = *(const float4*)(eps + off + N);
            su0 = __expf(0.5f * u[k0 + KSTEP + 2 * wp]);
            su1 = __expf(0.5f * u[k0 + KSTEP + 2 * wp + 1]);
        }

        // ---- batch fragment loads: 16x A-pairs + 8x B-pairs, all b64 ----
        v2f af[4][4];   // [kk][mi]
        v2f bf[4][2];   // [kk][ni]
        #pragma unroll
        for (int kk = 0; kk < 4; ++kk) {
            #pragma unroll
            for (int mi = 0; mi < 4; ++mi) {
                int row = wm * 64 + mi * 16 + la;
                af[kk][mi] = *(const v2f*)&As[row * ASTRIDE + kk * 4 + lh * 2];
            }
            #pragma unroll
            for (int ni = 0; ni < 2; ++ni) {
                int col = wn * 32 + ni * 16 + la;
                bf[kk][ni] = *(const v2f*)&Ws[(kk * 2 + lh) * 256 + col * 2];
            }
        }

        // ---- 32x v_wmma_f32_16x16x4_f32 ----
        #pragma unroll
        for (int kk = 0; kk < 4; ++kk)
            #pragma unroll
            for (int mi = 0; mi < 4; ++mi)
                #pragma unroll
                for (int ni = 0; ni < 2; ++ni)
                    acc[mi][ni] = __builtin_amdgcn_wmma_f32_16x16x4_f32(
                        false, af[kk][mi], false, bf[kk][ni], (short)0, acc[mi][ni],
                        false, false);
        __syncthreads();
    }

    // ---- epilogue: bias add (+ optional ReLU), C layout per ISA 7.12.2 ----
    #pragma unroll
    for (int mi = 0; mi < 4; ++mi) {
        #pragma unroll
        for (int ni = 0; ni < 2; ++ni) {
            int colL = wn * 32 + ni * 16 + la;
            float bv = bias[colL];
            size_t col = (size_t)(n0 + colL);
            #pragma unroll
            for (int r = 0; r < 8; ++r) {
                int row = m0 + wm * 64 + mi * 16 + (lh ? r + 8 : r);
                float val = acc[mi][ni][r] + bv;
                if (relu) val = fmaxf(val, 0.0f);
                out[(size_t)row * N + col] = val;
            }
        }
    }
}

extern "C" void kernel_launch(void* const* d_in, const int* in_sizes, int n_in,
                              void* d_out, int out_size, void* d_ws, size_t ws_size,
                              hipStream_t stream) {
    (void)in_sizes; (void)n_in; (void)out_size; (void)ws_size;
    const int B = 8192;
    const float* x = (const float*)d_in[0];
    float* h0 = (float*)d_ws;
    float* h1 = h0 + (size_t)B * 4096;

    struct { int K, N; } L[4] = { {2048, 4096}, {4096, 4096}, {4096, 4096}, {4096, 2048} };

    const float* a = x;
    for (int i = 0; i < 4; ++i) {
        const float* Mw = (const float*)d_in[1 + 4 * i];
        const float* u  = (const float*)d_in[2 + 4 * i];
        const float* v  = (const float*)d_in[3 + 4 * i];
        const float* ep = (const float*)d_in[4 + 4 * i];
        float* o = (i == 3) ? (float*)d_out : ((i == 1) ? h1 : h0);   // h0 -> h1 -> h0 -> out
        dim3 grid(L[i].N / TN, B / TM);
        mvg_layer_kernel<<<grid, 256, 0, stream>>>(a, Mw, u, v, ep, o,
                                                   L[i].K, L[i].N, (i < 3) ? 1 : 0);
        a = o;
    }
}